// LearnablePointFiltration_29454885716728
// MI455X (gfx1250) — compile-verified
//
#include <hip/hip_runtime.h>
#include <hip/hip_bf16.h>
#include <math.h>

typedef __attribute__((ext_vector_type(2))) float v2f;
typedef __attribute__((ext_vector_type(8))) float v8f;

#define N_PTS            8192
#define KNN              10
#define K1               (KNN + 1)              // keep 11: reference drops the smallest (self)
#define NEG_SLOPE        0.01f
#define ROWS_PER_WAVE    16
#define WAVES_PER_BLOCK  8
#define COL_SLICES       4                      // 4 waves share each 16-row group
#define ROW_GROUPS       (WAVES_PER_BLOCK / COL_SLICES)          // 2
#define ROWS_PER_BLOCK   (ROWS_PER_WAVE * ROW_GROUPS)            // 32
#define COLS_PER_SLICE   (N_PTS / COL_SLICES)                    // 2048
#define BIGF             1e30f

__device__ __forceinline__ float leaky(float x) { return x > 0.f ? x : NEG_SLOPE * x; }

// One 16-column tile: D = A x B + 0 where the spare K-slot carries the row norm:
// A row = (-2x, -2y, -2z, na), B col = (x', y', z', 1)  =>  D[M][N] = na - 2*dot.
// Squared distance = D + nb (single add). Insertion of x into each ascending
// top-11 list is pure v_med3_f32: new[k] = med3(x, old[k-1], old[k]), old[-1]=-inf.
// No self-exclusion here: like the reference, we keep the 11 smallest (self's
// ~0 squared distance ranks first) and drop rank 1 at the very end.
// tin/tout are distinct ping-pong buffers so the loop back-edge phi is copy-free.
__device__ __forceinline__ void process_tile(int jbase, int m, int hlf,
                                             const float4* __restrict__ ptsL,
                                             v2f avec,
                                             const float (&tin)[8][K1],
                                             float (&tout)[8][K1])
{
    float4 q = ptsL[jbase + m];
    v2f bvec;
    bvec[0] = hlf ? q.z : q.x;
    bvec[1] = hlf ? 1.0f : q.y;
    float nb = q.w;

    v8f acc = {};
    acc = __builtin_amdgcn_wmma_f32_16x16x4_f32(false, avec, false, bvec,
                                                (short)0, acc, false, false);

#pragma unroll
    for (int r = 0; r < 8; ++r) {
        float x = acc[r] + nb;                       // squared distance (may be ~-eps)
        tout[r][0] = __builtin_amdgcn_fmed3f(x, -BIGF, tin[r][0]);  // = min(x, t0)
#pragma unroll
        for (int k = 1; k < K1; ++k)
            tout[r][k] = __builtin_amdgcn_fmed3f(x, tin[r][k - 1], tin[r][k]);
    }
}

// Block = 32 rows x 8 waves: 2 row-groups x 4 column slices -> 256 blocks total.
// All 8192 points (xyz + |p|^2) cached in LDS. Top-k runs on SQUARED distances
// (sqrt monotone); per-lane lists live directly in the WMMA D layout (8 lists per
// lane, zero per-tile LDS traffic). Merge stage 1: per-wave 16-way tournament.
// Merge stage 2: 4-way tournament across column slices -> 11 smallest, drop the
// first (the self distance, exactly like the reference), sqrt, MLP.
__global__ __launch_bounds__(256, 1)
void knn_mlp_kernel(const float* __restrict__ pts,
                    const float* __restrict__ W1, const float* __restrict__ b1,
                    const float* __restrict__ W2, const float* __restrict__ b2,
                    const float* __restrict__ W3, const float* __restrict__ b3,
                    float* __restrict__ vf)
{
    __shared__ float4 ptsL[N_PTS];                          // 128 KB: xyz + squared norm
    __shared__ float  w1L[KNN * 64];
    __shared__ float  b1L[64];
    __shared__ float  w2L[64 * 64];
    __shared__ float  b2L[64];
    __shared__ float  w3L[64];
    __shared__ float  b3L;
    __shared__ float  mbuf[WAVES_PER_BLOCK][2][8][16][K1];  // 88 KB stage-1 lists
    __shared__ float  mbuf2[WAVES_PER_BLOCK][16][K1];       // 5.5 KB per-(row,slice) top-11
    __shared__ float  h1L[ROWS_PER_BLOCK][65];              //   8 KB padded hidden layer

    const int tid = threadIdx.x;

    // Cooperative fill of LDS: points (+norms) and MLP weights.
    for (int r = tid; r < N_PTS; r += 256) {
        float x = pts[3 * r + 0];
        float y = pts[3 * r + 1];
        float z = pts[3 * r + 2];
        ptsL[r] = make_float4(x, y, z, x * x + y * y + z * z);
    }
    for (int i = tid; i < KNN * 64; i += 256) w1L[i] = W1[i];
    for (int i = tid; i < 64 * 64; i += 256) w2L[i] = W2[i];
    if (tid < 64) { b1L[tid] = b1[tid]; b2L[tid] = b2[tid]; w3L[tid] = W3[tid]; }
    if (tid == 0) b3L = b3[0];
    __syncthreads();

    const int wave   = tid >> 5;
    const int lane   = tid & 31;
    const int hlf    = lane >> 4;                    // 16-lane half
    const int m      = lane & 15;
    const int rowgrp = wave >> 2;                    // which 16-row group (0..1)
    const int slice  = wave & 3;                     // which 2048-column slice (0..3)
    const int ibase  = blockIdx.x * ROWS_PER_BLOCK + rowgrp * ROWS_PER_WAVE;
    const int jstart = slice * COLS_PER_SLICE;
    const int jend   = jstart + COLS_PER_SLICE;

    // A tile (16x4 f32) held in registers: lanes 0-15 carry (K0,K1)=(-2x,-2y);
    // lanes 16-31 carry (K2,K3)=(-2z, na) -- the spare slot holds the row norm.
    v2f avec;
    {
        float4 p = ptsL[ibase + m];
        avec[0] = -2.f * (hlf ? p.z : p.x);
        avec[1] = hlf ? p.w : (-2.f * p.y);
    }

    float tA[8][K1], tB[8][K1];                      // ping-pong sorted top-11 lists
#pragma unroll
    for (int r = 0; r < 8; ++r)
#pragma unroll
        for (int k = 0; k < K1; ++k) tA[r][k] = BIGF;

    // 128 tiles per slice, processed in ping-pong pairs (copy-free phi).
    for (int jb = jstart; jb < jend; jb += 32) {
        process_tile(jb,      m, hlf, ptsL, avec, tA, tB);
        process_tile(jb + 16, m, hlf, ptsL, avec, tB, tA);
    }

    // Stage 1: dump per-lane lists, 16-way merge per (row, slice) inside the wave.
#pragma unroll
    for (int r = 0; r < 8; ++r)
#pragma unroll
        for (int k = 0; k < K1; ++k) mbuf[wave][hlf][r][m][k] = tA[r][k];
    __builtin_amdgcn_wave_barrier();
    asm volatile("s_wait_dscnt 0" ::: "memory");

    if (lane < 16) {
        const int h0 = lane >> 3, r0 = lane & 7;     // this lane merges row M = lane
        const float (*lists)[K1] = mbuf[wave][h0][r0];
        int hp[16];
#pragma unroll
        for (int n = 0; n < 16; ++n) hp[n] = 0;
#pragma unroll
        for (int k = 0; k < K1; ++k) {
            float best = BIGF;
            int bestn  = 0;
#pragma unroll
            for (int n = 0; n < 16; ++n) {
                float v0 = lists[n][hp[n]];
                bool  bl = v0 < best;
                best  = bl ? v0 : best;
                bestn = bl ? n : bestn;
            }
#pragma unroll
            for (int n = 0; n < 16; ++n) hp[n] += (bestn == n) ? 1 : 0;
            mbuf2[wave][lane][k] = best;             // still squared distance
        }
    }
    __syncthreads();

    // Stage 2: one thread per row merges the 4 column-slice lists; the first of
    // the 11 picks (the self distance) is discarded, matching the reference's
    // neg_top[:, 1:K+1]. Then sqrt + MLP.
    if (tid < ROWS_PER_BLOCK) {
        const int R = tid;                           // row within block
        const float* Ls[COL_SLICES];
#pragma unroll
        for (int s = 0; s < COL_SLICES; ++s)
            Ls[s] = mbuf2[(R >> 4) * COL_SLICES + s][R & 15];
        int hp[COL_SLICES];
#pragma unroll
        for (int s = 0; s < COL_SLICES; ++s) hp[s] = 0;

        float knnv[KNN];
#pragma unroll
        for (int k = 0; k < K1; ++k) {
            float best = BIGF;
            int bests  = 0;
#pragma unroll
            for (int s = 0; s < COL_SLICES; ++s) {
                float v0 = Ls[s][hp[s]];
                bool  bl = v0 < best;
                best  = bl ? v0 : best;
                bests = bl ? s : bests;
            }
#pragma unroll
            for (int s = 0; s < COL_SLICES; ++s) hp[s] += (bests == s) ? 1 : 0;
            if (k > 0)                               // drop rank-1 (self)
                knnv[k - 1] = (best > 0.f) ? __builtin_amdgcn_sqrtf(best) : 0.f;
        }

        // MLP: 10 -> 64 -> 64 -> 1 with leaky ReLU, weights broadcast from LDS.
        float* h1 = h1L[R];
        for (int o = 0; o < 64; ++o) {
            float acc1 = b1L[o];
#pragma unroll
            for (int i = 0; i < KNN; ++i) acc1 += knnv[i] * w1L[i * 64 + o];
            h1[o] = leaky(acc1);
        }
        float out = b3L;
        for (int o = 0; o < 64; ++o) {
            float acc2 = b2L[o];
            for (int i = 0; i < 64; ++i) acc2 += h1[i] * w2L[i * 64 + o];
            out += leaky(acc2) * w3L[o];
        }
        vf[blockIdx.x * ROWS_PER_BLOCK + R] = out;
    }
}

__global__ void edge_kernel(const float* __restrict__ pts,
                            const int* __restrict__ edges,
                            const float* __restrict__ vf,
                            float* __restrict__ ef, int n_edges)
{
    int e = blockIdx.x * blockDim.x + threadIdx.x;
    if (e >= n_edges) return;
    int u = edges[2 * e + 0];
    int v = edges[2 * e + 1];
    float dx = pts[3 * u + 0] - pts[3 * v + 0];
    float dy = pts[3 * u + 1] - pts[3 * v + 1];
    float dz = pts[3 * u + 2] - pts[3 * v + 2];
    float sq = dx * dx + dy * dy + dz * dz;
    float d  = (sq > 0.f) ? __builtin_amdgcn_sqrtf(sq) : 0.f;
    ef[e] = d + fmaxf(vf[u], vf[v]);
}

extern "C" void kernel_launch(void* const* d_in, const int* in_sizes, int n_in,
                              void* d_out, int out_size, void* d_ws, size_t ws_size,
                              hipStream_t stream)
{
    const float* pts   = (const float*)d_in[0];
    const int*   edges = (const int*)d_in[1];
    const float* W1    = (const float*)d_in[2];
    const float* b1    = (const float*)d_in[3];
    const float* W2    = (const float*)d_in[4];
    const float* b2    = (const float*)d_in[5];
    const float* W3    = (const float*)d_in[6];
    const float* b3    = (const float*)d_in[7];

    float* out = (float*)d_out;
    float* vf  = out;             // first 8192: vertex filtration values
    float* ef  = out + N_PTS;     // remaining: edge filtration values
    int n_edges = in_sizes[1] / 2;

    knn_mlp_kernel<<<N_PTS / ROWS_PER_BLOCK, 256, 0, stream>>>(
        pts, W1, b1, W2, b2, W3, b3, vf);
    edge_kernel<<<(n_edges + 255) / 256, 256, 0, stream>>>(
        pts, edges, vf, ef, n_edges);
}